// AbstractFullyConnected_70832600645725
// MI455X (gfx1250) — compile-verified
//
#include <hip/hip_runtime.h>

// ---------------------------------------------------------------------------
// DeepPoly MLP bound propagation on gfx1250 (MI455X).
// Heavy path: back-substitution GEMM chains in fp32 via V_WMMA_F32_16X16X4_F32.
// GEMM: 8 waves/WG, 64x64 WG tile, 16x32 per wave (2 accumulators).
// Staging: GLOBAL_LOAD_ASYNC_TO_LDS_B128 into double-buffered LDS, synced with
// s_wait_asynccnt + workgroup barriers; DMA of panel k+1 overlaps WMMA on k.
// ---------------------------------------------------------------------------

typedef __attribute__((ext_vector_type(2))) float v2f;
typedef __attribute__((ext_vector_type(8))) float v8f;

#define MEANC 0.1307f
#define STDC  0.3081f

#define TM 64   // output rows per workgroup (4 wave-rows of 16)
#define TN 64   // output cols per workgroup (2 wave-cols of 32)
#define TK 16   // K panel per LDS stage (4 wmma K-steps of 4)
#define APAD 4  // A row stride = TK+APAD = 20 floats: 16B-aligned rows,
                // stride-20 column reads hit 16 distinct LDS banks

// ---------------- normalization ----------------
__global__ __launch_bounds__(256) void k_normalize(
    const float* __restrict__ x, const float* __restrict__ lo,
    const float* __restrict__ hi, float* __restrict__ nx,
    float* __restrict__ nl, float* __restrict__ nh, int n) {
  int i = blockIdx.x * blockDim.x + threadIdx.x;
  if (i < n) {
    nx[i] = (x[i]  - MEANC) / STDC;
    nl[i] = (lo[i] - MEANC) / STDC;
    nh[i] = (hi[i] - MEANC) / STDC;
  }
}

// async DMA: per-lane 16B  global -> LDS  (tracked by ASYNCcnt)
__device__ __forceinline__ void async_b128(unsigned lds_off, unsigned gl_off,
                                           const float* base) {
  asm volatile("global_load_async_to_lds_b128 %0, %1, %2"
               :: "v"(lds_off), "v"(gl_off), "s"(base)
               : "memory");
}

// ---------------- fp32 WMMA GEMM: C[M,N] = A[M,K] @ B[K,N] ----------------
// Caller guarantees: K % 16 == 0, N % 16 == 0, 16B-aligned row strides.
// M/N raggedness handled by clamped loads + guarded stores.
__global__ __launch_bounds__(256) void k_gemm_wmma(
    const float* __restrict__ A, const float* __restrict__ B,
    float* __restrict__ C, int M, int N, int K) {
  __shared__ float As[2][TM][TK + APAD];
  __shared__ float Bs[2][TK][TN];

  const int tid  = threadIdx.x;
  const int lane = tid & 31;
  const int wave = tid >> 5;              // 0..7
  const int wr   = (wave >> 1) * 16;      // wave row offset in WG tile
  const int wc   = (wave & 1) * 32;       // wave col offset in WG tile
  const int row0 = blockIdx.y * TM;
  const int col0 = blockIdx.x * TN;

  // A-fragment layout (16x4 f32): lanes 0-15 -> M=lane, K={0,1};
  //                               lanes 16-31 -> M=lane-16, K={2,3}.
  const int frow = lane & 15;
  const int fkb  = (lane >> 4) * 2;

  // Staging assignment: one async float4 of A and of B per thread per panel.
  const int arow = tid >> 2;              // 0..63
  const int acol = (tid & 3) * 4;         // 0,4,8,12
  const int brow = tid >> 4;              // 0..15
  const int bcol = (tid & 15) * 4;        // 0..60

  // Clamp out-of-range rows/cols to valid addresses: duplicated data only
  // feeds output rows/cols that the store guard discards. Keeps every wave's
  // ASYNCcnt increment uniform (2 per panel) -> constant wait immediates.
  const int agr = (row0 + arow < M) ? (row0 + arow) : (M - 1);
  int bgc = col0 + bcol;
  if (bgc + 4 > N) bgc = N - 4;

  const unsigned aGl0 = ((unsigned)agr * (unsigned)K + (unsigned)acol) * 4u;
  const unsigned bGl0 = ((unsigned)brow * (unsigned)N + (unsigned)bgc) * 4u;
  const unsigned aStep = (unsigned)TK * 4u;               // per-panel A advance
  const unsigned bStep = (unsigned)TK * (unsigned)N * 4u; // per-panel B advance

  const unsigned aLds[2] = {
      (unsigned)(size_t)&As[0][arow][acol],
      (unsigned)(size_t)&As[1][arow][acol]};
  const unsigned bLds[2] = {
      (unsigned)(size_t)&Bs[0][brow][bcol],
      (unsigned)(size_t)&Bs[1][brow][bcol]};

  // kick off panel 0 into buffer 0
  async_b128(aLds[0], aGl0, A);
  async_b128(bLds[0], bGl0, B);

  v8f acc0 = {};
  v8f acc1 = {};

  int buf = 0;
  for (int kb = 0; kb < K; kb += TK) {
    const int kn = kb + TK;
    if (kn < K) {
      // DMA next panel into the other buffer; overlaps WMMA below
      async_b128(aLds[buf ^ 1], aGl0 + (unsigned)kn * 4u, A);
      async_b128(bLds[buf ^ 1], bGl0 + (unsigned)(kn / TK) * bStep, B);
      asm volatile("s_wait_asynccnt 2" ::: "memory");  // current panel landed
    } else {
      asm volatile("s_wait_asynccnt 0" ::: "memory");
    }
    __syncthreads();   // all waves' DMA for this panel visible in LDS

#pragma unroll
    for (int kk = 0; kk < TK; kk += 4) {
      v2f a, b0, b1;
      a.x  = As[buf][wr + frow][kk + fkb];
      a.y  = As[buf][wr + frow][kk + fkb + 1];
      b0.x = Bs[buf][kk + fkb][wc + frow];
      b0.y = Bs[buf][kk + fkb + 1][wc + frow];
      b1.x = Bs[buf][kk + fkb][wc + 16 + frow];
      b1.y = Bs[buf][kk + fkb + 1][wc + 16 + frow];
      acc0 = __builtin_amdgcn_wmma_f32_16x16x4_f32(
          false, a, false, b0, (short)0, acc0, false, false);
      acc1 = __builtin_amdgcn_wmma_f32_16x16x4_f32(
          false, a, false, b1, (short)0, acc1, false, false);
    }
    __syncthreads();   // everyone done reading buf before it is re-filled
    buf ^= 1;
  }

  // C/D layout: VGPR v -> lanes 0-15: M=v, N=lane; lanes 16-31: M=v+8, N=lane-16
  const int gc0   = col0 + wc + (lane & 15);
  const int gc1   = gc0 + 16;
  const int rbase = row0 + wr + ((lane >> 4) * 8);
#pragma unroll
  for (int v = 0; v < 8; ++v) {
    int gr = rbase + v;
    if (gr < M) {
      if (gc0 < N) C[(size_t)gr * N + gc0] = acc0[v];
      if (gc1 < N) C[(size_t)gr * N + gc1] = acc1[v];
    }
  }
}

// ---------------- block-wide reduction helper ----------------
__device__ __forceinline__ float block_reduce_256(float v) {
  __shared__ float sred[256];
  int t = threadIdx.x;
  sred[t] = v;
  __syncthreads();
  for (int s = 128; s > 0; s >>= 1) {
    if (t < s) sred[t] += sred[t + s];
    __syncthreads();
  }
  float r = sred[0];
  __syncthreads();
  return r;
}

// ---------------- y[r] = W[r,:] @ x + b[r] ----------------
__global__ __launch_bounds__(256) void k_gemv_bias(
    const float* __restrict__ W, const float* __restrict__ b,
    const float* __restrict__ x, float* __restrict__ y, int in) {
  int r = blockIdx.x;
  float acc = 0.0f;
  for (int c = threadIdx.x; c < in; c += 256)
    acc += W[(size_t)r * in + c] * x[c];
  float s = block_reduce_256(acc);
  if (threadIdx.x == 0) y[r] = s + b[r];
}

// ---------------- bias[r] += M[r,:] @ v ----------------
__global__ __launch_bounds__(256) void k_addmv(
    float* __restrict__ bias, const float* __restrict__ Mt,
    const float* __restrict__ v, int k) {
  int r = blockIdx.x;
  float acc = 0.0f;
  for (int c = threadIdx.x; c < k; c += 256)
    acc += Mt[(size_t)r * k + c] * v[c];
  float s = block_reduce_256(acc);
  if (threadIdx.x == 0) bias[r] += s;
}

// ---------------- ReLU back-substitution (in-place scale + bias) ----------------
// is_high=1: M' = where(M>0, M*wh, M*wl)*mask ; bias += (max(M,0)*mask) @ bhr
// is_high=0: M' = where(M>0, M*wl, M*wh)*mask ; bias += (min(M,0)*mask) @ bhr
__global__ __launch_bounds__(256) void k_relu_backsub(
    float* __restrict__ Mt, float* __restrict__ bias,
    const float* __restrict__ wl, const float* __restrict__ wh,
    const float* __restrict__ bhr, int k, int is_high) {
  int r = blockIdx.x;
  float acc = 0.0f;
  for (int c = threadIdx.x; c < k; c += 256) {
    size_t idx = (size_t)r * k + c;
    float m = Mt[idx];
    float whc = wh[c], wlc = wl[c];
    float mask = (whc != 0.0f) ? 1.0f : 0.0f;
    float scaled;
    if (is_high) {
      scaled = ((m > 0.0f) ? m * whc : m * wlc) * mask;
      acc += fmaxf(m, 0.0f) * mask * bhr[c];
    } else {
      scaled = ((m > 0.0f) ? m * wlc : m * whc) * mask;
      acc += fminf(m, 0.0f) * mask * bhr[c];
    }
    Mt[idx] = scaled;
  }
  float s = block_reduce_256(acc);
  if (threadIdx.x == 0) bias[r] += s;
}

// ---------------- interval matvec against input box ----------------
__global__ __launch_bounds__(256) void k_forward_boxes(
    const float* __restrict__ Mt, const float* __restrict__ bias,
    const float* __restrict__ nl, const float* __restrict__ nh,
    float* __restrict__ out, int k, int is_high) {
  int r = blockIdx.x;
  float acc = 0.0f;
  for (int c = threadIdx.x; c < k; c += 256) {
    float w = Mt[(size_t)r * k + c];
    float lo = nl[c], hi = nh[c];
    float pick = is_high ? ((w < 0.0f) ? lo : hi)
                         : ((w < 0.0f) ? hi : lo);
    acc += w * pick;
  }
  float s = block_reduce_256(acc);
  if (threadIdx.x == 0) out[r] = s + bias[r];
}

// ---------------- DeepPoly ReLU relaxation (records diag data) ----------------
__global__ __launch_bounds__(256) void k_abstract_relu(
    float* __restrict__ x, float* __restrict__ lo, float* __restrict__ hi,
    float* __restrict__ rwl, float* __restrict__ rwh, float* __restrict__ rbh,
    int n) {
  int i = blockIdx.x * blockDim.x + threadIdx.x;
  if (i >= n) return;
  float l = lo[i], h = hi[i];
  bool crossing = (l < 0.0f) && (h > 0.0f);
  float denom   = crossing ? (h - l) : 1.0f;
  float ub_slope = crossing ? (h / denom) : 0.0f;
  float ub_int   = crossing ? (-(l * h) / denom) : 0.0f;
  float lam  = (l * l > h * h) ? 0.0f : 1.0f;
  float keep = (h <= 0.0f) ? 0.0f : 1.0f;
  float w_high = crossing ? ub_slope : keep;
  float w_low  = crossing ? lam : keep;
  float b_high = ub_int;
  x[i]  = fmaxf(x[i], 0.0f);
  lo[i] = w_low * l;
  hi[i] = w_high * h + b_high;
  rwl[i] = w_low;
  rwh[i] = w_high;
  rbh[i] = b_high;
}

// ---------------------------------------------------------------------------
extern "C" void kernel_launch(void* const* d_in, const int* in_sizes, int n_in,
                              void* d_out, int out_size, void* d_ws, size_t ws_size,
                              hipStream_t stream) {
  (void)in_sizes; (void)n_in; (void)out_size; (void)ws_size;
  const int FCv[6] = {784, 2048, 2048, 2048, 2048, 10};

  const float* X  = (const float*)d_in[0];
  const float* LO = (const float*)d_in[1];
  const float* HI = (const float*)d_in[2];
  const float* W[5];
  const float* Bb[5];
  for (int i = 0; i < 5; ++i) {
    W[i]  = (const float*)d_in[3 + 2 * i];
    Bb[i] = (const float*)d_in[4 + 2 * i];
  }

  // ---- workspace carve-up (floats; every block is 16B-aligned) ----
  float* ws = (float*)d_ws;
  size_t off = 0;
  auto alloc = [&](size_t n) { float* p = ws + off; off += n; return p; };
  float* nx    = alloc(784);
  float* nl    = alloc(784);
  float* nh    = alloc(784);
  float* xA    = alloc(2048);
  float* xB    = alloc(2048);
  float* lowv  = alloc(2048);
  float* highv = alloc(2048);
  float* bh    = alloc(2048);
  float* bl    = alloc(2048);
  float* rwl[4]; float* rwh[4]; float* rbh[4];
  for (int r = 0; r < 4; ++r) {
    rwl[r] = alloc(2048); rwh[r] = alloc(2048); rbh[r] = alloc(2048);
  }
  const size_t MAT = (size_t)2048 * 2048;
  float* mbuf[4];
  for (int i = 0; i < 4; ++i) mbuf[i] = alloc(MAT);

  // ---- input normalization ----
  k_normalize<<<(784 + 255) / 256, 256, 0, stream>>>(X, LO, HI, nx, nl, nh, 784);

  float* xcur = xA;
  float* xnext = xB;

  for (int li = 0; li < 5; ++li) {
    const int ni = FCv[li + 1];

    // ================= back_sub over all records =================
    float* WhC = mbuf[0]; float* WhN = mbuf[1];
    float* WlC = mbuf[2]; float* WlN = mbuf[3];
    // identity @ W_li  ->  copy; identity @ b_li -> copy
    size_t wbytes = (size_t)ni * FCv[li] * sizeof(float);
    hipMemcpyAsync(WhC, W[li], wbytes, hipMemcpyDeviceToDevice, stream);
    hipMemcpyAsync(WlC, W[li], wbytes, hipMemcpyDeviceToDevice, stream);
    hipMemcpyAsync(bh, Bb[li], (size_t)ni * sizeof(float), hipMemcpyDeviceToDevice, stream);
    hipMemcpyAsync(bl, Bb[li], (size_t)ni * sizeof(float), hipMemcpyDeviceToDevice, stream);

    int curK = FCv[li];
    for (int j = li - 1; j >= 0; --j) {
      // ReLU record j (diag length == curK == FCv[j+1])
      k_relu_backsub<<<ni, 256, 0, stream>>>(WhC, bh, rwl[j], rwh[j], rbh[j], curK, 1);
      k_relu_backsub<<<ni, 256, 0, stream>>>(WlC, bl, rwl[j], rwh[j], rbh[j], curK, 0);
      // Linear record j: bias first (uses pre-GEMM matrix), then GEMM chain
      k_addmv<<<ni, 256, 0, stream>>>(bh, WhC, Bb[j], curK);
      k_addmv<<<ni, 256, 0, stream>>>(bl, WlC, Bb[j], curK);
      const int nj = FCv[j];
      dim3 grid((nj + TN - 1) / TN, (ni + TM - 1) / TM);
      k_gemm_wmma<<<grid, 256, 0, stream>>>(WhC, W[j], WhN, ni, nj, curK);
      k_gemm_wmma<<<grid, 256, 0, stream>>>(WlC, W[j], WlN, ni, nj, curK);
      float* t;
      t = WhC; WhC = WhN; WhN = t;
      t = WlC; WlC = WlN; WlN = t;
      curK = nj;
    }
    // concretize against the input box (curK == 784)
    k_forward_boxes<<<ni, 256, 0, stream>>>(WlC, bl, nl, nh, lowv, curK, 0);
    k_forward_boxes<<<ni, 256, 0, stream>>>(WhC, bh, nl, nh, highv, curK, 1);

    // ================= concrete forward =================
    const float* xv = (li == 0) ? nx : xcur;
    k_gemv_bias<<<ni, 256, 0, stream>>>(W[li], Bb[li], xv, xnext, FCv[li]);
    float* t = xcur; xcur = xnext; xnext = t;

    if (li < 4) {
      k_abstract_relu<<<(ni + 255) / 256, 256, 0, stream>>>(
          xcur, lowv, highv, rwl[li], rwh[li], rbh[li], ni);
    }
  }

  // outputs: x(10), low(10), high(10)
  float* out = (float*)d_out;
  hipMemcpyAsync(out,      xcur,  10 * sizeof(float), hipMemcpyDeviceToDevice, stream);
  hipMemcpyAsync(out + 10, lowv,  10 * sizeof(float), hipMemcpyDeviceToDevice, stream);
  hipMemcpyAsync(out + 20, highv, 10 * sizeof(float), hipMemcpyDeviceToDevice, stream);
}